// EarthMoverDistance_11012296147711
// MI455X (gfx1250) — compile-verified
//
#include <hip/hip_runtime.h>
#include <math.h>

// ---------------------------------------------------------------------------
// Entropic-OT Sinkhorn EMD for MI455X (gfx1250, wave32).
//
// C_ij = r1_i + r2_j - 2<p1_i,p2_j> (D=3) is never materialized: each 16x16
// tile of the Sinkhorn exponent (g_j - C_ij)/eps is produced by a single
// V_WMMA_F32_16X16X4_F32 with A pre-scaled by 2/eps and the accumulator
// preloaded with (g_j - r2_j)/eps - r1_i/eps.
//
// Round-2 change: the per-batch column working set (~40KB) is staged once per
// block into LDS in WMMA-fragment layout, so the 128-tile hot loop issues
// only ds_load_b64/b32 + v_wmma + v_exp — zero global traffic, no
// s_wait_loadcnt stalls (round-1 asm showed a full L2 round-trip per tile).
// ---------------------------------------------------------------------------

#define EPS_F      0.005f
#define INV_EPS_F  200.0f
#define MAX_ITER   50
#define B_         8
#define N_         2048
#define LOG_MU_F   (-7.6246189861593985f)   // -log(2048)
#define TILES      (N_ / 16)                 // 128 column tiles
#define WPB        4                         // waves per block (same batch)
#define PASS_GRID  (B_ * (N_ / 16) / WPB)    // 256 blocks

typedef __attribute__((ext_vector_type(2))) float v2f;
typedef __attribute__((ext_vector_type(8))) float v8f;

// ---- init: squared norms, zero potentials --------------------------------
__global__ void emd_init(const float* __restrict__ p1, const float* __restrict__ p2,
                         float* __restrict__ f, float* __restrict__ g,
                         float* __restrict__ r1, float* __restrict__ r2) {
    int i = blockIdx.x * blockDim.x + threadIdx.x;
    if (i < B_ * N_) {
        const float* a = p1 + (size_t)i * 3;
        const float* b = p2 + (size_t)i * 3;
        r1[i] = a[0]*a[0] + a[1]*a[1] + a[2]*a[2];
        r2[i] = b[0]*b[0] + b[1]*b[1] + b[2]*b[2];
        f[i] = 0.0f;
        g[i] = 0.0f;
    }
}

// ---- one Sinkhorn half-iteration: rowPotOut_i = eps*log_mu - eps*LSE_j ----
__global__ __launch_bounds__(128) void sinkhorn_pass(
    const float* __restrict__ Prow, const float* __restrict__ Pcol,
    const float* __restrict__ rowSq, const float* __restrict__ colSq,
    const float* __restrict__ colPot, float* __restrict__ rowPotOut) {

    // B-fragments per (col, half): half0 -> (x,y), half1 -> (z,0); plus the
    // pre-combined per-column affine term h = (g_j - r2_j)/eps.
    __shared__ float2 bfrag_lds[2 * N_];   // 32 KB
    __shared__ float  h_lds[N_];           //  8 KB

    const int lane    = threadIdx.x & 31;
    const int wave    = threadIdx.x >> 5;
    const int gw      = blockIdx.x * WPB + wave;
    const int batch   = (blockIdx.x * WPB) >> 7;   // shared by all waves in block
    const int rowBase = (gw & 127) * 16;
    const int l16     = lane & 15;
    const int half    = lane >> 4;
    const size_t bOff = (size_t)batch * N_;

    // cooperative staging of the whole column side for this batch
    for (int idx = threadIdx.x; idx < 2 * N_; idx += 128) {
        const int c  = idx >> 1;
        const int hf = idx & 1;
        const float* bp = Pcol + (bOff + c) * 3;
        float2 v;
        if (hf) { v.x = bp[2]; v.y = 0.0f; }
        else {
            v.x = bp[0]; v.y = bp[1];
            h_lds[c] = (colPot[bOff + c] - colSq[bOff + c]) * INV_EPS_F;
        }
        bfrag_lds[idx] = v;
    }

    // A fragment (16x4 f32): lanes 0-15 hold (K0,K1)=(x,y), lanes 16-31 (K2,K3)=(z,0).
    const float* ap = Prow + (bOff + rowBase + l16) * 3;
    v2f afrag;
    afrag.x = (half ? ap[2] : ap[0]) * (2.0f * INV_EPS_F);
    afrag.y = (half ? 0.0f  : ap[1]) * (2.0f * INV_EPS_F);

    float q[8];
#pragma unroll
    for (int k = 0; k < 8; ++k)
        q[k] = rowSq[bOff + rowBase + k + 8 * half] * INV_EPS_F;

    float m[8], s[8];
#pragma unroll
    for (int k = 0; k < 8; ++k) { m[k] = -__builtin_inff(); s[k] = 0.0f; }

    __syncthreads();

    for (int t = 0; t < TILES; ++t) {
        const int col = t * 16 + l16;
        const float2 bf = bfrag_lds[2 * col + half];   // ds_load_b64
        const float  h  = h_lds[col];                  // ds_load_b32
        v2f bfrag; bfrag.x = bf.x; bfrag.y = bf.y;

        v8f c;
#pragma unroll
        for (int k = 0; k < 8; ++k) c[k] = h - q[k];

        v8f d = __builtin_amdgcn_wmma_f32_16x16x4_f32(
            /*neg_a=*/false, afrag, /*neg_b=*/false, bfrag,
            /*c_mod=*/(short)0, c, /*reuse_a=*/false, /*reuse_b=*/false);

        // online log-sum-exp per slot over this lane's column subset
#pragma unroll
        for (int k = 0; k < 8; ++k) {
            const float v  = d[k];
            const float nm = fmaxf(m[k], v);
            s[k] = s[k] * __expf(m[k] - nm) + __expf(v - nm);
            m[k] = nm;
        }
    }

    // merge LSE state across the 16 lanes of each half (xor masks < 16 stay in-half)
#pragma unroll
    for (int k = 0; k < 8; ++k) {
#pragma unroll
        for (int off = 1; off < 16; off <<= 1) {
            const float om = __shfl_xor(m[k], off, 32);
            const float os = __shfl_xor(s[k], off, 32);
            const float nm = fmaxf(m[k], om);
            s[k] = s[k] * __expf(m[k] - nm) + os * __expf(om - nm);
            m[k] = nm;
        }
    }

    if (l16 == 0) {
#pragma unroll
        for (int k = 0; k < 8; ++k) {
            const float lse = m[k] + __logf(s[k]);
            rowPotOut[bOff + rowBase + k + 8 * half] = EPS_F * LOG_MU_F - EPS_F * lse;
        }
    }
}

// ---- final: dist_i = N * sum_j exp((f_i+g_j-C_ij)/eps) * C_ij; rowVal=sqrt --
__global__ __launch_bounds__(128) void emd_final(
    const float* __restrict__ P1, const float* __restrict__ P2,
    const float* __restrict__ r1, const float* __restrict__ r2,
    const float* __restrict__ f,  const float* __restrict__ g,
    float* __restrict__ rowVal) {

    __shared__ float2 bfrag_lds[2 * N_];   // 32 KB
    __shared__ float2 cg_lds[N_];          // 16 KB : (-r2_j, g_j)

    const int lane    = threadIdx.x & 31;
    const int wave    = threadIdx.x >> 5;
    const int gw      = blockIdx.x * WPB + wave;
    const int batch   = (blockIdx.x * WPB) >> 7;
    const int rowBase = (gw & 127) * 16;
    const int l16     = lane & 15;
    const int half    = lane >> 4;
    const size_t bOff = (size_t)batch * N_;

    for (int idx = threadIdx.x; idx < 2 * N_; idx += 128) {
        const int c  = idx >> 1;
        const int hf = idx & 1;
        const float* bp = P2 + (bOff + c) * 3;
        float2 v;
        if (hf) { v.x = bp[2]; v.y = 0.0f; }
        else {
            v.x = bp[0]; v.y = bp[1];
            float2 cg; cg.x = -r2[bOff + c]; cg.y = g[bOff + c];
            cg_lds[c] = cg;
        }
        bfrag_lds[idx] = v;
    }

    // A scaled by 2, Cacc = -(r1_i + r2_j)  ->  D = -C_ij
    const float* ap = P1 + (bOff + rowBase + l16) * 3;
    v2f afrag;
    afrag.x = (half ? ap[2] : ap[0]) * 2.0f;
    afrag.y = (half ? 0.0f  : ap[1]) * 2.0f;

    float qr[8], fu[8], acc[8];
#pragma unroll
    for (int k = 0; k < 8; ++k) {
        qr[k]  = r1[bOff + rowBase + k + 8 * half];
        fu[k]  = f [bOff + rowBase + k + 8 * half];
        acc[k] = 0.0f;
    }

    __syncthreads();

    for (int t = 0; t < TILES; ++t) {
        const int col = t * 16 + l16;
        const float2 bf = bfrag_lds[2 * col + half];
        const float2 cg = cg_lds[col];
        v2f bfrag; bfrag.x = bf.x; bfrag.y = bf.y;

        v8f c;
#pragma unroll
        for (int k = 0; k < 8; ++k) c[k] = cg.x - qr[k];

        v8f d = __builtin_amdgcn_wmma_f32_16x16x4_f32(
            false, afrag, false, bfrag, (short)0, c, false, false);

#pragma unroll
        for (int k = 0; k < 8; ++k) {
            const float Cv = -d[k];                                  // C_ij
            const float tt = (fu[k] + cg.y + d[k]) * INV_EPS_F;      // (f+g-C)/eps
            acc[k] = fmaf(__expf(tt), Cv, acc[k]);
        }
    }

#pragma unroll
    for (int k = 0; k < 8; ++k) {
#pragma unroll
        for (int off = 1; off < 16; off <<= 1)
            acc[k] += __shfl_xor(acc[k], off, 32);
    }

    if (l16 == 0) {
#pragma unroll
        for (int k = 0; k < 8; ++k) {
            const float dist = (float)N_ * acc[k];
            rowVal[bOff + rowBase + k + 8 * half] = sqrtf(dist + 1e-12f);
        }
    }
}

// ---- deterministic mean over B*N row values (single block, fixed order) ---
__global__ void reduce_mean(const float* __restrict__ rowVal, float* __restrict__ out) {
    __shared__ float sm[256];
    float sum = 0.0f;
    for (int i = threadIdx.x; i < B_ * N_; i += 256) sum += rowVal[i];
    sm[threadIdx.x] = sum;
    __syncthreads();
    for (int st = 128; st > 0; st >>= 1) {
        if ((int)threadIdx.x < st) sm[threadIdx.x] += sm[threadIdx.x + st];
        __syncthreads();
    }
    if (threadIdx.x == 0) out[0] = sm[0] / (float)(B_ * N_);
}

extern "C" void kernel_launch(void* const* d_in, const int* in_sizes, int n_in,
                              void* d_out, int out_size, void* d_ws, size_t ws_size,
                              hipStream_t stream) {
    const float* p1 = (const float*)d_in[0];   // [8,2048,3] f32
    const float* p2 = (const float*)d_in[1];   // [8,2048,3] f32
    float* ws = (float*)d_ws;                  // 5 * 64KB = 320KB used
    float* f      = ws;
    float* g      = ws + 1 * B_ * N_;
    float* r1     = ws + 2 * B_ * N_;
    float* r2     = ws + 3 * B_ * N_;
    float* rowVal = ws + 4 * B_ * N_;

    emd_init<<<(B_ * N_ + 255) / 256, 256, 0, stream>>>(p1, p2, f, g, r1, r2);

    for (int it = 0; it < MAX_ITER; ++it) {
        // f-update: rows = pcs1, cols = pcs2, reads g
        sinkhorn_pass<<<PASS_GRID, 128, 0, stream>>>(p1, p2, r1, r2, g, f);
        // g-update: transposed problem (C symmetric in roles), reads f
        sinkhorn_pass<<<PASS_GRID, 128, 0, stream>>>(p2, p1, r2, r1, f, g);
    }

    emd_final<<<PASS_GRID, 128, 0, stream>>>(p1, p2, r1, r2, f, g, rowVal);
    reduce_mean<<<1, 256, 0, stream>>>(rowVal, (float*)d_out);
}